// PenaltyMethod_68427418959978
// MI455X (gfx1250) — compile-verified
//
#include <hip/hip_runtime.h>
#include <hip/hip_bf16.h>
#include <stdint.h>

// Problem constants from the reference
#define B_  4
#define P_  8
#define V_  2048
#define E_  2048
#define F_  4096
#define BIGF 1e10f

#define SPLITV 8            // split V across blocks for occupancy
#define TPB 256             // 8 wave32 per block
#define WAVES (TPB / 32)
#define VT_TOTAL (V_ / 16)              // 128 v-tiles
#define VT_PER_BLOCK (VT_TOTAL / SPLITV)   // 16
#define VT_PER_WAVE (VT_PER_BLOCK / WAVES) // 2
#define ET (E_ / 16)                    // 128 e-tiles

typedef __attribute__((ext_vector_type(2))) float v2f;
typedef __attribute__((ext_vector_type(8))) float v8f;

// lane-xor add via ds_swizzle (group-of-32: offset = xor<<10 | and=0x1f)
#define SWZ_MIN(x, imm) __builtin_fminf((x), __int_as_float(__builtin_amdgcn_ds_swizzle(__float_as_int(x), (imm))))
#define SWZ_ADD(x, imm) ((x) + __int_as_float(__builtin_amdgcn_ds_swizzle(__float_as_int(x), (imm))))

// ---------------------------------------------------------------------------
// Chamfer partial: one block = (b, p, v-slice). Pairwise term via
// V_WMMA_F32_16X16X4_F32 with K augmented to [x, y, |em|^2, 0] x [-2px,-2py,1,0].
// ---------------------------------------------------------------------------
__global__ __launch_bounds__(TPB) void chamfer_partial_kernel(
    const float* __restrict__ xs, const float* __restrict__ pm,
    const float* __restrict__ em, const int* __restrict__ lens,
    const uint8_t* __restrict__ bmask, float2* __restrict__ partial)
{
  __shared__ float4 ldsA4[E_];      // {x, y, |em|^2, 0} per edge point (32 KB)
  __shared__ float  ldsR[WAVES * 2];

  const int blk = blockIdx.x;                 // b*P*SPLITV + p*SPLITV + s
  const int s   = blk % SPLITV;
  const int p   = (blk / SPLITV) % P_;
  const int b   = blk / (SPLITV * P_);
  const int tid = threadIdx.x;

  // Stage transformed edgemaps into LDS; invalid rows -> [0,0,BIG,0] so the
  // WMMA output is BIG regardless of v (mask folded into the matmul).
  const int len = lens[b * P_ + p];
  const float* embp = em + ((size_t)(b * P_ + p)) * E_ * 2;
  for (int e = tid; e < E_; e += TPB) {
    float ex = embp[e * 2 + 0];
    float ey = embp[e * 2 + 1];
    bool valid = e < len;
    float4 rec;
    rec.x = valid ? ex : 0.0f;
    rec.y = valid ? ey : 0.0f;
    rec.z = valid ? (ex * ex + ey * ey) : BIGF;
    rec.w = 0.0f;
    ldsA4[e] = rec;
  }
  __syncthreads();

  const int  lane = tid & 31;
  const int  wave = tid >> 5;
  const int  ln   = lane & 15;
  const bool hi   = lane >= 16;

  // projection matrix row-major 3x4 for this p (uniform per block)
  const float* pmp = pm + p * 12;
  const float m00 = pmp[0], m01 = pmp[1], m02 = pmp[2],  m03 = pmp[3];
  const float m10 = pmp[4], m11 = pmp[5], m12 = pmp[6],  m13 = pmp[7];
  const float m20 = pmp[8], m21 = pmp[9], m22 = pmp[10], m23 = pmp[11];

  const float* lbase = reinterpret_cast<const float*>(ldsA4);
  const int foff = ln * 4 + (hi ? 2 : 0);   // A-matrix lane layout (M=ln, K pair)

  float num = 0.0f, den = 0.0f;

  for (int i = 0; i < VT_PER_WAVE; ++i) {
    const int vt = s * VT_PER_BLOCK + wave * VT_PER_WAVE + i;
    const int v  = vt * 16 + ln;

    // project vertex v (lanes 0-15 and 16-31 both compute; only low half used)
    const float* xp = xs + ((size_t)b * V_ + v) * 3;
    const float x = xp[0], y = xp[1], z = xp[2];
    const float w  = m20 * x + m21 * y + m22 * z + m23;
    const float rw = 1.0f / w;
    const float px = (m00 * x + m01 * y + m02 * z + m03) * rw;
    const float py = (m10 * x + m11 * y + m12 * z + m13) * rw;

    // B matrix 4x16: lanes 0-15 -> K=0,1 = (-2px,-2py); lanes 16-31 -> K=2,3 = (1,0)
    v2f bmat;
    bmat.x = hi ? 1.0f : -2.0f * px;
    bmat.y = hi ? 0.0f : -2.0f * py;

    const v8f zero = {};
    v8f acc;
#pragma unroll
    for (int k = 0; k < 8; ++k) acc[k] = BIGF;

    // Inner loop: ds_load_b64 + v_wmma_f32_16x16x4_f32 + 8x v_min
    for (int et = 0; et < ET; ++et) {
      v2f a = *reinterpret_cast<const v2f*>(lbase + et * 64 + foff);
      v8f d = __builtin_amdgcn_wmma_f32_16x16x4_f32(
          /*neg_a=*/false, a, /*neg_b=*/false, bmat,
          /*c_mod=*/(short)0, zero, /*reuse_a=*/false, /*reuse_b=*/false);
#pragma unroll
      for (int k = 0; k < 8; ++k) acc[k] = __builtin_fminf(acc[k], d[k]);
    }

    // reduce 8 e-rows in-lane, then e-rows 8-15 across the half-wave (SWAPX16)
    float m = acc[0];
#pragma unroll
    for (int k = 1; k < 8; ++k) m = __builtin_fminf(m, acc[k]);
    m = SWZ_MIN(m, 0x401f);

    if (!hi) {
      const float p2 = px * px + py * py;          // add |pts|^2 after the min
      const float msk = bmask[((size_t)(b * P_ + p)) * V_ + v] ? 1.0f : 0.0f;
      num += (m + p2) * msk;
      den += msk;
    }
  }

  // wave32 xor-tree (lanes 16-31 carry zeros), then cross-wave via LDS
  num = SWZ_ADD(num, 0x041f); den = SWZ_ADD(den, 0x041f);
  num = SWZ_ADD(num, 0x081f); den = SWZ_ADD(den, 0x081f);
  num = SWZ_ADD(num, 0x101f); den = SWZ_ADD(den, 0x101f);
  num = SWZ_ADD(num, 0x201f); den = SWZ_ADD(den, 0x201f);
  num = SWZ_ADD(num, 0x401f); den = SWZ_ADD(den, 0x401f);

  if (lane == 0) { ldsR[wave * 2] = num; ldsR[wave * 2 + 1] = den; }
  __syncthreads();
  if (tid == 0) {
    float n = 0.0f, d = 0.0f;
    for (int wv = 0; wv < WAVES; ++wv) { n += ldsR[wv * 2]; d += ldsR[wv * 2 + 1]; }
    partial[blk] = make_float2(n, d);
  }
}

// ---------------------------------------------------------------------------
// Finalize chamfer: deterministic combine of the 8 v-slices per (b,p),
// then mean over P. One wave does everything (partials are 2 KB).
// ---------------------------------------------------------------------------
__global__ void chamfer_finalize_kernel(const float2* __restrict__ partial,
                                        float* __restrict__ out)
{
  __shared__ float pp[B_ * P_];
  const int t = threadIdx.x;
  if (t < B_ * P_) {
    float n = 0.0f, d = 0.0f;
    for (int s2 = 0; s2 < SPLITV; ++s2) {
      float2 v = partial[t * SPLITV + s2];
      n += v.x; d += v.y;
    }
    pp[t] = n / fmaxf(d, 1.0f);
  }
  __syncthreads();
  if (t < B_) {
    float acc = 0.0f;
    for (int p = 0; p < P_; ++p) acc += pp[t * P_ + p];
    out[t] = acc / (float)P_;
  }
}

// ---------------------------------------------------------------------------
// Volume constraint: one block per batch, LDS tree reduction (deterministic).
// ---------------------------------------------------------------------------
__global__ __launch_bounds__(256) void volume_kernel(
    const float* __restrict__ xs, const int* __restrict__ faces,
    const float* __restrict__ target, float* __restrict__ out)
{
  __shared__ float red[256];
  const int b = blockIdx.x, tid = threadIdx.x;
  const int*   fb = faces + (size_t)b * F_ * 3;
  const float* xb = xs + (size_t)b * V_ * 3;

  float local = 0.0f;
  for (int f = tid; f < F_; f += 256) {
    const int i0 = fb[f * 3 + 0], i1 = fb[f * 3 + 1], i2 = fb[f * 3 + 2];
    const float* a = xb + (size_t)i0 * 3;
    const float* c = xb + (size_t)i1 * 3;
    const float* d = xb + (size_t)i2 * 3;
    const float ax = a[0], ay = a[1], az = a[2];
    const float bx = c[0], by = c[1], bz = c[2];
    const float cx = d[0], cy = d[1], cz = d[2];
    const float crx = ay * bz - az * by;
    const float cry = az * bx - ax * bz;
    const float crz = ax * by - ay * bx;
    local += (crx * cx + cry * cy + crz * cz) * (1.0f / 6.0f);
  }
  red[tid] = local;
  __syncthreads();
  for (int stride = 128; stride > 0; stride >>= 1) {
    if (tid < stride) red[tid] += red[tid + stride];
    __syncthreads();
  }
  if (tid == 0) {
    const float vols = fabsf(red[0]);
    const float dv = vols - target[b];
    out[B_ + b] = dv * dv;
  }
}

// ---------------------------------------------------------------------------
extern "C" void kernel_launch(void* const* d_in, const int* in_sizes, int n_in,
                              void* d_out, int out_size, void* d_ws, size_t ws_size,
                              hipStream_t stream) {
  const float*   xs    = (const float*)d_in[0];   // (B,V,3)
  const float*   pm    = (const float*)d_in[1];   // (P,3,4)
  const float*   em    = (const float*)d_in[2];   // (B,P,E,2)
  const int*     lens  = (const int*)d_in[3];     // (B,P)
  const uint8_t* bmask = (const uint8_t*)d_in[4]; // (B,P,V) bool
  const int*     faces = (const int*)d_in[5];     // (B,F,3)
  const float*   tv    = (const float*)d_in[6];   // (B,)
  float*  out     = (float*)d_out;                // [chamfer(4) | vol_error(4)]
  float2* partial = (float2*)d_ws;                // B*P*SPLITV float2 = 2 KB

  chamfer_partial_kernel<<<B_ * P_ * SPLITV, TPB, 0, stream>>>(xs, pm, em, lens,
                                                               bmask, partial);
  chamfer_finalize_kernel<<<1, 32, 0, stream>>>(partial, out);
  volume_kernel<<<B_, 256, 0, stream>>>(xs, faces, tv, out);
}